// LossFunction_50517405335656
// MI455X (gfx1250) — compile-verified
//
#include <hip/hip_runtime.h>
#include <math.h>

#define N_PRED   131072
#define N_GT     256
#define N_CLS    80
#define IOU_TH   0.5f
#define NBLK     (N_PRED / 256)   // 512 blocks of 256 threads

typedef __attribute__((ext_vector_type(2))) float v2f;
typedef __attribute__((ext_vector_type(8))) float v8f;

// ctrl[0] = gi invalidated last iteration (-1 = none)
// ctrl[1] = done flag (global max dropped below threshold)
// ctrl[2] = number of valid matches

__device__ __forceinline__ float iou1(float4 p, float4 g) {
    float iw = fminf(p.z, g.z) - fmaxf(p.x, g.x);
    float ih = fminf(p.w, g.w) - fmaxf(p.y, g.y);
    iw = fmaxf(iw, 0.0f);
    ih = fmaxf(ih, 0.0f);
    float inter = iw * ih;
    float ap = (p.z - p.x) * (p.w - p.y);
    float ag = (g.z - g.x) * (g.w - g.y);
    float un = ap + ag - inter;
    return inter / fmaxf(un, 1e-9f);
}

// Stage one 16B gt box per lane into LDS via the CDNA5 async-to-LDS path
// (GLOBAL_LOAD_ASYNC_TO_LDS_B128, GVS mode: per-lane LDS address from vdst, global address saddr+vaddr).
// Tracked by ASYNCcnt; caller must s_wait_asynccnt 0 before a barrier/use.
__device__ __forceinline__ void async_stage_gt(const float4* gt, float4* gts_lds, int tid) {
    unsigned lds_off = (unsigned)(size_t)(&gts_lds[0]) + (unsigned)tid * 16u; // low 32b of generic = LDS offset
    unsigned goff    = (unsigned)tid * 16u;
    unsigned long long base = (unsigned long long)(size_t)gt;
    asm volatile("global_load_async_to_lds_b128 %0, %1, %2"
                 :: "v"(lds_off), "v"(goff), "s"(base)
                 : "memory");
}
__device__ __forceinline__ void wait_async() {
    asm volatile("s_wait_asynccnt 0" ::: "memory");
}

// ---------- Phase 1: per-row IoU max/argmax over all 256 gt (gt staged in LDS) ----------
__global__ void k_init(const float4* __restrict__ pred, const float4* __restrict__ gt,
                       float* __restrict__ rowmax, int* __restrict__ rowarg,
                       int* __restrict__ colinv, int* __restrict__ ctrl) {
    __shared__ float4 gts[N_GT];
    int tid = threadIdx.x;
    async_stage_gt(gt, gts, tid);            // DMA gt tile into LDS (no VGPR round-trip)
    if (blockIdx.x == 0) {
        colinv[tid] = 0;
        if (tid == 0) { ctrl[0] = -1; ctrl[1] = 0; ctrl[2] = 0; }
    }
    int row = blockIdx.x * 256 + tid;
    float4 p = pred[row];                    // overlaps with async LDS fill
    wait_async();
    __syncthreads();
    float best = -1.0f; int barg = 0;
#pragma unroll 8
    for (int g = 0; g < N_GT; ++g) {
        float v = iou1(p, gts[g]);
        if (v > best) { best = v; barg = g; }   // strict > keeps smallest index on ties
    }
    rowmax[row] = best;
    rowarg[row] = barg;
}

// ---------- Phase 2a: repair rows whose argmax column was just invalidated, then partial argmax ----------
__global__ void k_scan(const float4* __restrict__ pred, const float4* __restrict__ gt,
                       float* __restrict__ rowmax, int* __restrict__ rowarg,
                       const int* __restrict__ colinv, const int* __restrict__ ctrl,
                       float* __restrict__ pval, int* __restrict__ pidx) {
    __shared__ float4 gts[N_GT];
    __shared__ int    cinv[N_GT];
    __shared__ float  sval[256];
    __shared__ int    sidx[256];
    int tid = threadIdx.x;
    if (ctrl[1]) {                       // matching finished: nothing will be selected
        if (tid == 0) { pval[blockIdx.x] = -2.0f; pidx[blockIdx.x] = 0; }
        return;
    }
    async_stage_gt(gt, gts, tid);        // async DMA of gt boxes overlaps the loads below
    cinv[tid] = colinv[tid];
    int gi_last = ctrl[0];
    int row = blockIdx.x * 256 + tid;
    float v = rowmax[row];
    int need = (gi_last >= 0 && v > -1.5f && rowarg[row] == gi_last);
    wait_async();
    __syncthreads();
    if (need) {
        float4 p = pred[row];
        float best = -1.0f; int barg = 0;
#pragma unroll 4
        for (int g = 0; g < N_GT; ++g) {
            if (cinv[g]) continue;
            float x = iou1(p, gts[g]);
            if (x > best) { best = x; barg = g; }
        }
        rowmax[row] = best; rowarg[row] = barg;
        v = best;
    }
    sval[tid] = v; sidx[tid] = row;
    __syncthreads();
    for (int s = 128; s > 0; s >>= 1) {
        if (tid < s) {
            float a = sval[tid], b = sval[tid + s];
            int   ib = sidx[tid + s];
            if (b > a || (b == a && ib < sidx[tid])) { sval[tid] = b; sidx[tid] = ib; }
        }
        __syncthreads();
    }
    if (tid == 0) { pval[blockIdx.x] = sval[0]; pidx[blockIdx.x] = sidx[0]; }
}

// ---------- Phase 2b: final argmax over 512 partials; select match, invalidate row+col ----------
__global__ void k_pick(const float* __restrict__ pval, const int* __restrict__ pidx,
                       float* __restrict__ rowmax, const int* __restrict__ rowarg,
                       int* __restrict__ colinv, int* __restrict__ ctrl,
                       int* __restrict__ mpi, int* __restrict__ mgi,
                       const float* __restrict__ cls) {
    __shared__ float sval[NBLK];
    __shared__ int   sidx[NBLK];
    int tid = threadIdx.x;                     // 512 threads
    if (ctrl[1]) return;
    sval[tid] = pval[tid]; sidx[tid] = pidx[tid];
    __syncthreads();
    for (int s = NBLK / 2; s > 0; s >>= 1) {
        if (tid < s) {
            float a = sval[tid], b = sval[tid + s];
            int   ib = sidx[tid + s];
            if (b > a || (b == a && ib < sidx[tid])) { sval[tid] = b; sidx[tid] = ib; }
        }
        __syncthreads();
    }
    if (tid == 0) {
        float v = sval[0];
        int   pi = sidx[0];
        if (v >= IOU_TH) {
            int gi = rowarg[pi];
            int k  = ctrl[2];
            mpi[k] = pi; mgi[k] = gi;
            ctrl[2] = k + 1;
            rowmax[pi] = -2.0f;                // row out of play
            colinv[gi] = 1;                    // column out of play
            ctrl[0] = gi;
            __builtin_prefetch(cls + (size_t)pi * N_CLS, 0, 0);  // warm logits row (global_prefetch_b8)
        } else {
            ctrl[1] = 1; ctrl[0] = -1;         // all remaining maxima < thresh -> all invalid
        }
    }
}

// ---------- Phase 3: CE + SmoothL1 over matches; totals reduced via V_WMMA_F32_16X16X4_F32 ----------
__global__ void k_loss(const float4* __restrict__ pred, const float4* __restrict__ gt,
                       const float* __restrict__ cls, const int* __restrict__ gtc,
                       const int* __restrict__ ctrl, const int* __restrict__ mpi,
                       const int* __restrict__ mgi, float* __restrict__ out) {
    __shared__ float ce_s[N_GT];
    __shared__ float sl_s[N_GT];
    int tid = threadIdx.x;                     // 256 threads = 8 waves (wave32)
    ce_s[tid] = 0.0f;
    sl_s[tid] = 0.0f;
    __syncthreads();
    int nm   = ctrl[2];
    int wave = tid >> 5;
    int lane = tid & 31;
    for (int k = wave; k < nm; k += 8) {       // one wave per match
        int pi = mpi[k];
        int gi = mgi[k];
        const float* lrow = cls + (size_t)pi * N_CLS;
        float v0 = lrow[lane];                              // 0..31   < 80
        float v1 = lrow[lane + 32];                         // 32..63  < 80
        float v2 = (lane < N_CLS - 64) ? lrow[lane + 64] : -3.0e38f;  // 64..79
        float m = fmaxf(v0, fmaxf(v1, v2));
        for (int off = 16; off > 0; off >>= 1) m = fmaxf(m, __shfl_xor(m, off));
        float s = expf(v0 - m) + expf(v1 - m) +
                  ((lane < N_CLS - 64) ? expf(v2 - m) : 0.0f);
        for (int off = 16; off > 0; off >>= 1) s += __shfl_xor(s, off);
        if (lane == 0) {
            int tgt = gtc[gi];
            ce_s[k] = (m + logf(s)) - lrow[tgt];           // -log_softmax[target]
            float4 p = pred[pi];
            float4 g = gt[gi];
            float d0 = p.x - g.x, d1 = p.y - g.y, d2 = p.z - g.z, d3 = p.w - g.w;
            float a0 = fabsf(d0), a1 = fabsf(d1), a2 = fabsf(d2), a3 = fabsf(d3);
            float s0 = (a0 < 1.0f) ? 0.5f * d0 * d0 : a0 - 0.5f;
            float s1 = (a1 < 1.0f) ? 0.5f * d1 * d1 : a1 - 0.5f;
            float s2 = (a2 < 1.0f) ? 0.5f * d2 * d2 : a2 - 0.5f;
            float s3 = (a3 < 1.0f) ? 0.5f * d3 * d3 : a3 - 0.5f;
            sl_s[k] = 0.25f * (s0 + s1 + s2 + s3);
        }
    }
    __syncthreads();
    if (tid < 32) {
        // Sum 256 padded values with D = A(16x4) x ones(4x16) + C accumulation.
        // With B = all-ones the (M,K) slot packing is sum-invariant, so arbitrary
        // (lane, vgpr) packing of each 64-element chunk is exact f32 accumulation.
        v2f ones; ones[0] = 1.0f; ones[1] = 1.0f;
        v8f acc_ce = {0.f, 0.f, 0.f, 0.f, 0.f, 0.f, 0.f, 0.f};
        v8f acc_sl = {0.f, 0.f, 0.f, 0.f, 0.f, 0.f, 0.f, 0.f};
#pragma unroll
        for (int c = 0; c < 4; ++c) {
            v2f a, b;
            a[0] = ce_s[c * 64 + lane]; a[1] = ce_s[c * 64 + 32 + lane];
            b[0] = sl_s[c * 64 + lane]; b[1] = sl_s[c * 64 + 32 + lane];
            acc_ce = __builtin_amdgcn_wmma_f32_16x16x4_f32(
                false, a, false, ones, (short)0, acc_ce, false, false);
            acc_sl = __builtin_amdgcn_wmma_f32_16x16x4_f32(
                false, b, false, ones, (short)0, acc_sl, false, false);
        }
        float pce = 0.0f, psl = 0.0f;
#pragma unroll
        for (int i = 0; i < 8; ++i) { pce += acc_ce[i]; psl += acc_sl[i]; }
        // D[m][n] = S_m replicated over n; lane0 holds S_0..7, lane16 holds S_8..15
        float ce_tot = __shfl(pce, 0) + __shfl(pce, 16);
        float sl_tot = __shfl(psl, 0) + __shfl(psl, 16);
        if (lane == 0) {
            float denom = fmaxf((float)nm, 1.0f);
            out[0] = (nm > 0) ? ce_tot / denom : 0.0f;   // LAMBDA_CLS = 1
            out[1] = (nm > 0) ? sl_tot / denom : 0.0f;   // LAMBDA_BOX = 1
        }
    }
}

extern "C" void kernel_launch(void* const* d_in, const int* in_sizes, int n_in,
                              void* d_out, int out_size, void* d_ws, size_t ws_size,
                              hipStream_t stream) {
    const float4* pred = (const float4*)d_in[0];   // [131072,4] f32
    const float4* gt   = (const float4*)d_in[1];   // [256,4]    f32
    const float*  cls  = (const float*)d_in[2];    // [131072,80] f32
    const int*    gtc  = (const int*)d_in[3];      // [256] i32
    float* out = (float*)d_out;                    // 2 floats

    char* w = (char*)d_ws;
    float* rowmax = (float*)w;  w += (size_t)N_PRED * 4;
    int*   rowarg = (int*)w;    w += (size_t)N_PRED * 4;
    float* pval   = (float*)w;  w += (size_t)NBLK * 4;
    int*   pidx   = (int*)w;    w += (size_t)NBLK * 4;
    int*   colinv = (int*)w;    w += (size_t)N_GT * 4;
    int*   ctrl   = (int*)w;    w += 64;
    int*   mpi    = (int*)w;    w += (size_t)N_GT * 4;
    int*   mgi    = (int*)w;    w += (size_t)N_GT * 4;
    // total ~1.05 MB of d_ws

    k_init<<<NBLK, 256, 0, stream>>>(pred, gt, rowmax, rowarg, colinv, ctrl);
    for (int it = 0; it < N_GT; ++it) {
        k_scan<<<NBLK, 256, 0, stream>>>(pred, gt, rowmax, rowarg, colinv, ctrl, pval, pidx);
        k_pick<<<1, NBLK, 0, stream>>>(pval, pidx, rowmax, rowarg, colinv, ctrl, mpi, mgi, cls);
    }
    k_loss<<<1, 256, 0, stream>>>(pred, gt, cls, gtc, ctrl, mpi, mgi, out);
}